// FoldLayer_79809082294662
// MI455X (gfx1250) — compile-verified
//
#include <hip/hip_runtime.h>

// Fold / col2im overlap-add, gather formulation, strip-per-wave.
// patches: (B=16, GH=64, GW=64, 9*128) f32, kernel-major rows of C floats.
// out:     (B=16, 128, 128, 128) f32 ('same' padding cropped).
//
// Memory-bound: 302 MB read + 134 MB written => ~19 us floor @ 23.3 TB/s.
// One wave32 handles 8 consecutive output pixels (4 odd/even pairs) at fixed
// (b, oy): vertical contributors hoisted, horizontal contributors become two
// adds per pair. All contributor rows for the strip are staged via CDNA5
// async global->LDS B128 copies (ASYNCcnt), one wait, then summed in the
// reference's (i,j)-lexicographic order (bit-exact) and stored non-temporally.

typedef float v4f __attribute__((ext_vector_type(4)));

namespace {
constexpr int Bn    = 16;
constexpr int GH    = 64;
constexpr int GW    = 64;
constexpr int C     = 128;
constexpr int PADc  = 1;
constexpr int OUT_H = 128;
constexpr int OUT_W = 128;
constexpr int C4    = C / 4;            // 32 float4 per channel row
constexpr int PAIRS = 4;                // pixel pairs per wave strip
constexpr int STRIP = 2 * PAIRS;        // 8 output pixels per wave
constexpr int ROWS_PER_PAIR = 6;        // max: 2 vertical x {A, B0, B2}
constexpr int ROWS_PER_WAVE = PAIRS * ROWS_PER_PAIR;   // 24 rows = 12 KB
constexpr int WAVES_PER_BLOCK = 4;      // 128 threads, 48 KB LDS/block
constexpr int STRIPS_PER_ROW = OUT_W / STRIP;          // 16
}

__global__ __launch_bounds__(WAVES_PER_BLOCK * 32)
void fold_overlap_add_kernel(const v4f* __restrict__ p4, v4f* __restrict__ out4)
{
    __shared__ __align__(16) v4f stage[WAVES_PER_BLOCK * ROWS_PER_WAVE * 32];

    const int lane = threadIdx.x & 31;
    const int wave = threadIdx.x >> 5;
    const int sid  = (blockIdx.x * WAVES_PER_BLOCK * 32 + threadIdx.x) >> 5;

    const int t0 = (sid & (STRIPS_PER_ROW - 1)) * PAIRS;   // first patch col t
    const int oy = (sid >> 4) & (OUT_H - 1);
    const int b  = sid >> 11;

    // Vertical contributors for py = oy + 1, in ascending-i order
    // (matches the reference's i-major accumulation order).
    const int py = oy + PADc;
    int yv[2], iv[2], nv = 0;
    if (py & 1) {
        yv[0] = py >> 1; iv[0] = 1; nv = 1;
    } else {
        const int y = py >> 1;
        if (y < GH) { yv[nv] = y; iv[nv] = 0; ++nv; }
        yv[nv] = y - 1; iv[nv] = 2; ++nv;      // y-1 >= 0 for all even py >= 2
    }
    // Hoisted per-contributor bases (patch-row units of C floats).
    int rowy[2], kb[2];
    for (int v = 0; v < nv; ++v) {
        rowy[v] = (b * GH + yv[v]) * GW;
        kb[v]   = iv[v] * 3;
    }

    v4f* wslot = &stage[wave * ROWS_PER_WAVE * 32];
    const unsigned lds0 = (unsigned)(unsigned long long)wslot; // flat->LDS offset

    // ---- Issue phase: async global->LDS copies for the whole strip --------
    // Pair p covers ox = 2t, 2t+1 with t = t0 + p:
    //   pixel A (px = 2t+1, odd):  row (x=t,   j=1)
    //   pixel B (px = 2t+2, even): row (x=t+1, j=0) [if t+1 < GW], (x=t, j=2)
    #pragma unroll
    for (int p = 0; p < PAIRS; ++p) {
        const int  t      = t0 + p;
        const bool haveB0 = (t + 1 < GW);
        for (int v = 0; v < nv; ++v) {
            const int rowbase = (rowy[v] + t) * 9 + kb[v];       // patch rows
            const unsigned d  = lds0 + (unsigned)(((p * 2 + v) * 3) * 32 + lane) * 16u;

            const v4f* srcA = p4 + (long)(rowbase + 1) * C4 + lane;       // (t, j=1)
            asm volatile("global_load_async_to_lds_b128 %0, %1, off"
                         :: "v"(d), "v"(srcA) : "memory");
            if (haveB0) {
                const v4f* srcB0 = p4 + (long)(rowbase + 9) * C4 + lane;  // (t+1, j=0)
                asm volatile("global_load_async_to_lds_b128 %0, %1, off"
                             :: "v"(d + 512u), "v"(srcB0) : "memory");
            }
            const v4f* srcB2 = p4 + (long)(rowbase + 2) * C4 + lane;      // (t, j=2)
            asm volatile("global_load_async_to_lds_b128 %0, %1, off"
                         :: "v"(d + 1024u), "v"(srcB2) : "memory");
        }
    }

    // Wait for all async transfers of this wave.
    asm volatile("s_wait_asynccnt 0x0" ::: "memory");

    // ---- Consume phase: sum in (i,j)-lexicographic order, NT store --------
    const int obase0 = ((b * OUT_H + oy) * OUT_W + 2 * t0) * C4 + lane;
    #pragma unroll
    for (int p = 0; p < PAIRS; ++p) {
        const int  t      = t0 + p;
        const bool haveB0 = (t + 1 < GW);
        v4f accA = (v4f)0.0f;
        v4f accB = (v4f)0.0f;
        for (int v = 0; v < nv; ++v) {
            const int r = (p * 2 + v) * 3;
            accA += wslot[(r + 0) * 32 + lane];                  // (iv, j=1)
            if (haveB0) accB += wslot[(r + 1) * 32 + lane];      // (iv, j=0)
            accB += wslot[(r + 2) * 32 + lane];                  // (iv, j=2)
        }
        // Output is write-once, never re-read: non-temporal B128 stores.
        __builtin_nontemporal_store(accA, &out4[obase0 + (2 * p)     * C4]);
        __builtin_nontemporal_store(accB, &out4[obase0 + (2 * p + 1) * C4]);
    }
}

extern "C" void kernel_launch(void* const* d_in, const int* in_sizes, int n_in,
                              void* d_out, int out_size, void* d_ws, size_t ws_size,
                              hipStream_t stream)
{
    (void)in_sizes; (void)n_in; (void)d_ws; (void)ws_size; (void)out_size;

    const v4f* p4  = (const v4f*)d_in[0];
    v4f*       out = (v4f*)d_out;

    // 16*128*16 strips, one wave32 each, 4 waves per block.
    const int total_waves = Bn * OUT_H * STRIPS_PER_ROW;   // 32768
    const int blocks = total_waves / WAVES_PER_BLOCK;      // 8192

    hipLaunchKernelGGL(fold_overlap_add_kernel, dim3(blocks),
                       dim3(WAVES_PER_BLOCK * 32), 0, stream, p4, out);
}